// Pooling_Layer_77369540870266
// MI455X (gfx1250) — compile-verified
//
#include <hip/hip_runtime.h>
#include <hip/hip_bf16.h>

typedef __attribute__((ext_vector_type(2))) float v2f;
typedef __attribute__((ext_vector_type(8))) float v8f;

#define C_IN   64
#define C_OUT  128
#define MAXN   32
#define ROWS_PER_BLOCK 64
#define AS_LD  68   // 64 + 4 pad: keeps 16B alignment, bank stride 4
#define BS_LD  129  // 128 + 1 pad: de-conflicts transpose scatter writes

// ---------------------------------------------------------------------------
// Kernel 1: x[b,p,o] = sum_i W[o,i] * in[b,p,i]   (M x 64 @ 64 x 128, fp32)
// One block = 8 waves, 64 rows. Wave w owns columns [16w, 16w+16).
// Uses V_WMMA_F32_16X16X4_F32, 16 k-steps of 4.
// ---------------------------------------------------------------------------
__global__ __launch_bounds__(256) void pool_gemm_wmma(
    const float* __restrict__ in,   // [B][rows][64]
    const float* __restrict__ W,    // [128][64]
    float* __restrict__ x,          // [B][rows][128]
    int rows)
{
    __shared__ float As[ROWS_PER_BLOCK][AS_LD];   // ~17 KB
    __shared__ float Bs[C_IN][BS_LD];             // ~33 KB (W transposed: Bs[k][n]=W[n][k])

    const int b        = blockIdx.y;
    const int row_base = blockIdx.x * ROWS_PER_BLOCK;
    const int tid      = threadIdx.x;
    const float* inb   = in + (size_t)b * rows * C_IN;

    // Stage A tile: 64x64 f32 = 1024 float4, 4 per thread. Clamp OOB rows.
    #pragma unroll
    for (int i = tid; i < ROWS_PER_BLOCK * (C_IN / 4); i += 256) {
        const int r  = i >> 4;          // 0..63
        const int c4 = i & 15;          // 0..15
        int gr = row_base + r;
        if (gr >= rows) gr = rows - 1;  // clamped read; store side predicates
        const float4 v = *(const float4*)(inb + (size_t)gr * C_IN + c4 * 4);
        *(float4*)(&As[r][c4 * 4]) = v;
    }
    // Stage W transposed: Bs[k][n] = W[n][k]; 2048 float4 reads, scatter to LDS.
    #pragma unroll
    for (int i = tid; i < C_OUT * (C_IN / 4); i += 256) {
        const int n  = i >> 4;          // 0..127
        const int k4 = i & 15;          // 0..15
        const float4 v = *(const float4*)(W + n * C_IN + k4 * 4);
        Bs[k4 * 4 + 0][n] = v.x;
        Bs[k4 * 4 + 1][n] = v.y;
        Bs[k4 * 4 + 2][n] = v.z;
        Bs[k4 * 4 + 3][n] = v.w;
    }
    __syncthreads();

    const int lane    = tid & 31;
    const int wave    = tid >> 5;       // 0..7 -> 16-column slice
    const int colbase = wave * 16;
    const int hl      = lane >> 4;      // half-wave: K offset +0 / +2
    const int l16     = lane & 15;

    // Preload all 16 B fragments (4x16 K-major per ISA layout) into registers.
    v2f bfrag[16];
    #pragma unroll
    for (int kk = 0; kk < 16; ++kk) {
        const int kb = kk * 4 + hl * 2;
        v2f bf;
        bf.x = Bs[kb + 0][colbase + l16];
        bf.y = Bs[kb + 1][colbase + l16];
        bfrag[kk] = bf;
    }

    #pragma unroll
    for (int rt = 0; rt < 4; ++rt) {
        v8f acc = {};
        #pragma unroll
        for (int kk = 0; kk < 16; ++kk) {
            const int kb = kk * 4 + hl * 2;
            v2f af;
            af.x = As[rt * 16 + l16][kb + 0];
            af.y = As[rt * 16 + l16][kb + 1];
            acc = __builtin_amdgcn_wmma_f32_16x16x4_f32(
                false, af, false, bfrag[kk], (short)0, acc, false, false);
        }
        // C/D layout: VGPR r -> M = rt*16 + r + 8*hl, N = colbase + l16
        #pragma unroll
        for (int r = 0; r < 8; ++r) {
            const int gm = row_base + rt * 16 + r + 8 * hl;
            if (gm < rows) {
                x[((size_t)b * rows + gm) * C_OUT + colbase + l16] = acc[r];
            }
        }
    }
}

// ---------------------------------------------------------------------------
// Kernel 2: out[b,p,:] = sum_m w[p,m] * x[b, id[p,m], :],
//           w[p,m] = |pn[p,m]|*mask[p,m] / (sum_m' ... + 1e-8)
// One wave32 per (b,p). Lane m owns neighbor m for the weight reduction;
// lane l owns channels [4l, 4l+4) (float4) for the gather-accumulate.
// x is L2-resident (164 MB < 192 MB), so gathers hit L2, not HBM.
// ---------------------------------------------------------------------------
__global__ __launch_bounds__(256) void pool_gather(
    const float* __restrict__ x,     // [B][rows][128]
    const int*   __restrict__ ids,   // [OUT_PN][32]
    const float* __restrict__ mask,  // [OUT_PN][32]
    const float* __restrict__ pn,    // [OUT_PN][32]
    float* __restrict__ out,         // [B][OUT_PN][128]
    int rows, int out_pn, int nbatch)
{
    const int lane = threadIdx.x & 31;
    const int gw   = blockIdx.x * 8 + (threadIdx.x >> 5);
    const int total = nbatch * out_pn;
    if (gw >= total) return;                     // uniform per wave
    const int b = gw / out_pn;
    const int p = gw % out_pn;

    // lane <-> neighbor for weight computation
    const float pm  = fabsf(pn[p * MAXN + lane]) * mask[p * MAXN + lane];
    const int   idm = ids[p * MAXN + lane];
    float s = pm;
    #pragma unroll
    for (int off = 16; off > 0; off >>= 1) s += __shfl_xor(s, off, 32);
    const float wl = pm / (s + 1e-8f);

    const float* xb = x + (size_t)b * rows * C_OUT;
    float4 acc = {0.f, 0.f, 0.f, 0.f};
    #pragma unroll
    for (int m = 0; m < MAXN; ++m) {
        const float wm = __shfl(wl, m, 32);
        const int   im = __shfl(idm, m, 32);
        const float4 v = *(const float4*)(xb + (size_t)im * C_OUT + lane * 4);
        acc.x += wm * v.x;
        acc.y += wm * v.y;
        acc.z += wm * v.z;
        acc.w += wm * v.w;
    }
    *(float4*)(out + ((size_t)(b * out_pn + p)) * C_OUT + lane * 4) = acc;
}

extern "C" void kernel_launch(void* const* d_in, const int* in_sizes, int n_in,
                              void* d_out, int out_size, void* d_ws, size_t ws_size,
                              hipStream_t stream)
{
    const float* in_pc = (const float*)d_in[0];   // [B][rows][64]
    const int*   ids   = (const int*)d_in[1];     // [OUT_PN][32]
    const float* mask  = (const float*)d_in[2];   // [OUT_PN][32]
    const float* pn    = (const float*)d_in[3];   // [OUT_PN][32]
    const float* W     = (const float*)d_in[4];   // [128][64]

    const int out_pn = in_sizes[1] / MAXN;                    // 10000
    const int nbatch = out_size / (out_pn * C_OUT);           // 8
    const int rows   = in_sizes[0] / (nbatch * C_IN);         // 40001

    float* x = (float*)d_ws;  // intermediate [B][rows][128] (~164 MB, L2-resident)

    dim3 g1((rows + ROWS_PER_BLOCK - 1) / ROWS_PER_BLOCK, nbatch);
    pool_gemm_wmma<<<g1, 256, 0, stream>>>(in_pc, W, x, rows);

    const int total_waves = nbatch * out_pn;
    const int g2 = (total_waves + 7) / 8;
    pool_gather<<<g2, 256, 0, stream>>>(x, ids, mask, pn, (float*)d_out,
                                        rows, out_pn, nbatch);
}